// GAT_65764539236885
// MI455X (gfx1250) — compile-verified
//
#include <hip/hip_runtime.h>
#include <hip/hip_bf16.h>
#include <math.h>

typedef __attribute__((ext_vector_type(2))) float v2f;
typedef __attribute__((ext_vector_type(8))) float v8f;

#define NN_IN_CH 64
#define NN_HID 16
#define NN_HEADS 4
#define NN_F1 (NN_HEADS * NN_HID)   // 64
#define NEG_SLOPE 0.2f
#define GAT_EPS 1e-16f

// ---------------------------------------------------------------------------
// Layer-1 GEMM: h1[N,64] = x[N,64] @ W1[64,64] using V_WMMA_F32_16X16X4_F32.
// One wave computes a 16x64 output strip (4 accumulator tiles), K=64 in 16
// steps of 4.  fp32 WMMA keeps full precision through the attention softmax.
// Layouts per CDNA5 ISA 7.12.2:
//   A 16x4:  lane = M (both halves), VGPR0 = K{0|2}, VGPR1 = K{1|3}
//   B 4x16:  lane = N (both halves), VGPR0 = K{0|2}, VGPR1 = K{1|3}
//   C/D:     VGPR r -> M = r + 8*half, N = lane&15
// ---------------------------------------------------------------------------
__global__ void __launch_bounds__(256)
gat_gemm1_wmma(const float* __restrict__ x, const float* __restrict__ W1,
               float* __restrict__ h1, int n)
{
    int wave = (blockIdx.x * blockDim.x + threadIdx.x) >> 5;
    int lane = threadIdx.x & 31;
    int row_base = wave * 16;
    if (row_base >= n) return;                // wave-uniform: EXEC stays all-1s
    int half = lane >> 4;
    int lid  = lane & 15;

    const float* xr = x + (size_t)(row_base + lid) * NN_IN_CH;

    v8f acc0 = {}, acc1 = {}, acc2 = {}, acc3 = {};
    for (int k = 0; k < NN_IN_CH; k += 4) {
        v2f a;
        a.x = xr[k + 2 * half + 0];
        a.y = xr[k + 2 * half + 1];
        const float* wr0 = W1 + (size_t)(k + 2 * half + 0) * NN_F1;
        const float* wr1 = W1 + (size_t)(k + 2 * half + 1) * NN_F1;
        v2f b0; b0.x = wr0[lid +  0]; b0.y = wr1[lid +  0];
        v2f b1; b1.x = wr0[lid + 16]; b1.y = wr1[lid + 16];
        v2f b2; b2.x = wr0[lid + 32]; b2.y = wr1[lid + 32];
        v2f b3; b3.x = wr0[lid + 48]; b3.y = wr1[lid + 48];
        acc0 = __builtin_amdgcn_wmma_f32_16x16x4_f32(false, a, false, b0, (short)0, acc0, false, false);
        acc1 = __builtin_amdgcn_wmma_f32_16x16x4_f32(false, a, false, b1, (short)0, acc1, false, false);
        acc2 = __builtin_amdgcn_wmma_f32_16x16x4_f32(false, a, false, b2, (short)0, acc2, false, false);
        acc3 = __builtin_amdgcn_wmma_f32_16x16x4_f32(false, a, false, b3, (short)0, acc3, false, false);
    }

    float* outp = h1 + (size_t)(row_base + 8 * half) * NN_F1 + lid;
#pragma unroll
    for (int r = 0; r < 8; ++r) {
        outp[(size_t)r * NN_F1 +  0] = acc0[r];
        outp[(size_t)r * NN_F1 + 16] = acc1[r];
        outp[(size_t)r * NN_F1 + 32] = acc2[r];
        outp[(size_t)r * NN_F1 + 48] = acc3[r];
    }
}

// a_src1[n,h] / a_dst1[n,h]: per-(node,head) 16-wide dots with att vectors.
__global__ void gat_attn1(const float* __restrict__ h1,
                          const float* __restrict__ att_src,
                          const float* __restrict__ att_dst,
                          float* __restrict__ asrc, float* __restrict__ adst, int n)
{
    int t = blockIdx.x * blockDim.x + threadIdx.x;
    if (t >= n * NN_HEADS) return;
    int node = t >> 2, hd = t & 3;
    const float* hp = h1 + (size_t)node * NN_F1 + hd * NN_HID;
    const float* as = att_src + hd * NN_HID;
    const float* ad = att_dst + hd * NN_HID;
    float s = 0.f, d = 0.f;
#pragma unroll
    for (int c = 0; c < NN_HID; ++c) { float v = hp[c]; s += v * as[c]; d += v * ad[c]; }
    asrc[t] = s;
    adst[t] = d;
}

// Init accumulators: out1 <- b1 (bias folded in), d1/d2 <- 0, d_out <- b2.
__global__ void gat_init(float* __restrict__ d1, float* __restrict__ out1,
                         float* __restrict__ d2, float* __restrict__ outf,
                         const float* __restrict__ b1, const float* __restrict__ b2, int n)
{
    int t = blockIdx.x * blockDim.x + threadIdx.x;
    if (t < n * NN_F1) out1[t] = b1[t & (NN_F1 - 1)];
    if (t < n * NN_HEADS) d1[t] = 0.f;
    if (t < n) { d2[t] = 0.f; outf[t] = b2[0]; }
}

__device__ __forceinline__ void edge_sd(const long long* __restrict__ ei,
                                        int E, int e, int& s, int& d)
{
    if (e < E) { s = (int)ei[e]; d = (int)ei[(size_t)E + e]; }
    else       { s = d = e - E; }                 // synthesized self-loop
}

__device__ __forceinline__ float lrelu(float a) { return a >= 0.f ? a : a * NEG_SLOPE; }

// Segment softmax denominator, layer 1 (no max-subtraction: logits bounded,
// e/denom invariant to the shift).
__global__ void gat_edge_denom1(const long long* __restrict__ ei, int E, int ET,
                                const float* __restrict__ asrc,
                                const float* __restrict__ adst,
                                float* __restrict__ d1)
{
    int e = blockIdx.x * blockDim.x + threadIdx.x;
    if (e >= ET) return;
    int s, d; edge_sd(ei, E, e, s, d);
    const float* as = asrc + (size_t)s * NN_HEADS;
    const float* ad = adst + (size_t)d * NN_HEADS;
#pragma unroll
    for (int hd = 0; hd < NN_HEADS; ++hd)
        atomicAdd(&d1[(size_t)d * NN_HEADS + hd], __expf(lrelu(as[hd] + ad[hd])));
}

// Weighted message aggregation, layer 1: 16 lanes per edge, each lane owns a
// float4 channel group (head = group>>2).  256B coalesced gather of h1[src].
__global__ void gat_edge_agg1(const long long* __restrict__ ei, int E, int ET,
                              const float* __restrict__ asrc,
                              const float* __restrict__ adst,
                              const float* __restrict__ d1,
                              const float* __restrict__ h1,
                              float* __restrict__ out1)
{
    long long t = (long long)blockIdx.x * blockDim.x + threadIdx.x;
    if (t >= (long long)ET * 16) return;
    int e = (int)(t >> 4), g = (int)(t & 15);
    int s, d; edge_sd(ei, E, e, s, d);
    int hd = g >> 2;
    float a = asrc[(size_t)s * NN_HEADS + hd] + adst[(size_t)d * NN_HEADS + hd];
    float w = __expf(lrelu(a)) / (d1[(size_t)d * NN_HEADS + hd] + GAT_EPS);
    float4 hv = ((const float4*)(h1 + (size_t)s * NN_F1))[g];
    float* op = out1 + (size_t)d * NN_F1 + g * 4;
    atomicAdd(op + 0, hv.x * w);
    atomicAdd(op + 1, hv.y * w);
    atomicAdd(op + 2, hv.z * w);
    atomicAdd(op + 3, hv.w * w);
}

// Per-node: ELU then dot with W2[64,1] -> hh[n].  One wave32 per node.
__global__ void __launch_bounds__(256)
gat_finalize1(const float* __restrict__ out1, const float* __restrict__ W2,
              float* __restrict__ hh, int n)
{
    int wave = (blockIdx.x * blockDim.x + threadIdx.x) >> 5;
    int lane = threadIdx.x & 31;
    if (wave >= n) return;
    float acc = 0.f;
#pragma unroll
    for (int i = 0; i < 2; ++i) {
        int ch = lane + i * 32;
        float v = out1[(size_t)wave * NN_F1 + ch];
        v = v > 0.f ? v : expm1f(v);              // ELU
        acc += v * W2[ch];
    }
#pragma unroll
    for (int off = 16; off > 0; off >>= 1)
        acc += __shfl_xor(acc, off, 32);
    if (lane == 0) hh[wave] = acc;
}

// Layer-2 softmax denominator (1 head, 1 channel; attention from hh on the fly).
__global__ void gat_edge_denom2(const long long* __restrict__ ei, int E, int ET,
                                const float* __restrict__ hh,
                                const float* __restrict__ as2,
                                const float* __restrict__ ad2,
                                float* __restrict__ d2)
{
    int e = blockIdx.x * blockDim.x + threadIdx.x;
    if (e >= ET) return;
    int s, d; edge_sd(ei, E, e, s, d);
    float a = hh[s] * as2[0] + hh[d] * ad2[0];
    atomicAdd(&d2[d], __expf(lrelu(a)));
}

// Layer-2 aggregation straight into d_out (pre-initialized with b2).
__global__ void gat_edge_agg2(const long long* __restrict__ ei, int E, int ET,
                              const float* __restrict__ hh,
                              const float* __restrict__ as2,
                              const float* __restrict__ ad2,
                              const float* __restrict__ d2,
                              float* __restrict__ outf)
{
    int e = blockIdx.x * blockDim.x + threadIdx.x;
    if (e >= ET) return;
    int s, d; edge_sd(ei, E, e, s, d);
    float a = hh[s] * as2[0] + hh[d] * ad2[0];
    float w = __expf(lrelu(a)) / (d2[d] + GAT_EPS);
    atomicAdd(&outf[d], hh[s] * w);
}

extern "C" void kernel_launch(void* const* d_in, const int* in_sizes, int n_in,
                              void* d_out, int out_size, void* d_ws, size_t ws_size,
                              hipStream_t stream)
{
    const float*     x   = (const float*)d_in[0];
    const long long* ei  = (const long long*)d_in[1];   // int64 edge_index [2,E]
    const float*     W1  = (const float*)d_in[2];
    const float*     as1 = (const float*)d_in[3];
    const float*     ad1 = (const float*)d_in[4];
    const float*     b1  = (const float*)d_in[5];
    const float*     W2  = (const float*)d_in[6];
    const float*     as2 = (const float*)d_in[7];
    const float*     ad2 = (const float*)d_in[8];
    const float*     b2  = (const float*)d_in[9];
    float*           out = (float*)d_out;

    const int n  = in_sizes[0] / NN_IN_CH;   // 100000
    const int E  = in_sizes[1] / 2;          // 1600000
    const int ET = E + n;                    // + self-loops

    // Carve scratch (~57 MB) from d_ws, 256B aligned.
    char*  ws  = (char*)d_ws;
    size_t off = 0;
    auto carve = [&](size_t bytes) -> float* {
        float* p = (float*)(ws + off);
        off += (bytes + 255) & ~(size_t)255;
        return p;
    };
    float* h1    = carve((size_t)n * NN_F1 * sizeof(float));
    float* out1  = carve((size_t)n * NN_F1 * sizeof(float));
    float* asrc1 = carve((size_t)n * NN_HEADS * sizeof(float));
    float* adst1 = carve((size_t)n * NN_HEADS * sizeof(float));
    float* d1    = carve((size_t)n * NN_HEADS * sizeof(float));
    float* hh    = carve((size_t)n * sizeof(float));
    float* d2    = carve((size_t)n * sizeof(float));
    (void)ws_size; (void)n_in; (void)out_size;

    const int blk = 256;
    int gemm_waves = (n + 15) / 16;
    gat_gemm1_wmma<<<(gemm_waves * 32 + blk - 1) / blk, blk, 0, stream>>>(x, W1, h1, n);
    gat_attn1<<<(n * NN_HEADS + blk - 1) / blk, blk, 0, stream>>>(h1, as1, ad1, asrc1, adst1, n);
    gat_init<<<((long long)n * NN_F1 + blk - 1) / blk, blk, 0, stream>>>(d1, out1, d2, out, b1, b2, n);
    gat_edge_denom1<<<(ET + blk - 1) / blk, blk, 0, stream>>>(ei, E, ET, asrc1, adst1, d1);
    gat_edge_agg1<<<(unsigned)(((long long)ET * 16 + blk - 1) / blk), blk, 0, stream>>>(
        ei, E, ET, asrc1, adst1, d1, h1, out1);
    gat_finalize1<<<(n * 32 + blk - 1) / blk, blk, 0, stream>>>(out1, W2, hh, n);
    gat_edge_denom2<<<(ET + blk - 1) / blk, blk, 0, stream>>>(ei, E, ET, hh, as2, ad2, d2);
    gat_edge_agg2<<<(ET + blk - 1) / blk, blk, 0, stream>>>(ei, E, ET, hh, as2, ad2, d2, out);
}